// GGNN_60266981097696
// MI455X (gfx1250) — compile-verified
//
#include <hip/hip_runtime.h>

#define NN 10000       // nodes = 625 * 16 = (125 blocks * 5 waves) * 16  -- exact
#define NE 320000      // edges = 40000 * 8
#define DIM 256        // feature dim = 16 * 16
#define LAYERS 5
#define KC 64          // K-chunk staged in LDS for the GRU kernel
#define BSTRIDE 68     // 64 + 4 pad words: lane l hits banks {4l,4l+1}, half-wave
                       // (+2 words) hits {4l+2,4l+3} -> all 64 banks distinct

typedef __attribute__((ext_vector_type(2))) float v2f;
typedef __attribute__((ext_vector_type(8))) float v8f;

// D = A(16x4) * B(4x16) + C, fp32 WMMA (V_WMMA_F32_16X16X4_F32)
__device__ __forceinline__ v8f wmma_f32(v2f a, v2f b, v8f c) {
  return __builtin_amdgcn_wmma_f32_16x16x4_f32(
      /*neg_a=*/false, a, /*neg_b=*/false, b,
      /*c_mod=*/(short)0, c, /*reuse_a=*/false, /*reuse_b=*/false);
}

// Async Global->LDS copy, 16 bytes, tracked on ASYNCcnt (ISA 08 §4:
// GLOBAL_LOAD_ASYNC_TO_LDS_B128, GV mode: VDST = LDS byte address VGPR).
__device__ __forceinline__ void async_copy_b128(void* lds_dst, const float* src) {
  uint32_t loff = (uint32_t)(uintptr_t)lds_dst;   // LDS aperture: addr[31:0] = offset
  asm volatile("global_load_async_to_lds_b128 %0, %1, off"
               :: "v"(loff), "v"(src) : "memory");
}
__device__ __forceinline__ void async_wait0() {
  asm volatile("s_wait_asynccnt 0" ::: "memory");
}

// ---------------------------------------------------------------------------
// m = h @ W.  5 waves/block share one 16-col B panel staged in LDS (16 KB).
// Frag layouts (ISA 7.12.2): A lane<16 -> {A[l,k],A[l,k+1]}, lane>=16 -> k+2,k+3
//                            B lane<16 -> {B[k,l],B[k+1,l]}, lane>=16 -> k+2,k+3
//                            C reg r   -> lane<16: C[r, l], lane>=16: C[8+r, l]
// ---------------------------------------------------------------------------
__global__ __launch_bounds__(160) void gemm_nn(const float* __restrict__ A,
                                               const float* __restrict__ B,
                                               float* __restrict__ C) {
  __shared__ __align__(16) float Bs[DIM * 16];   // Bs[k*16 + c] = B[k][jbase+c]
  const int tid = threadIdx.x;
  const int wid = tid >> 5;                      // 0..4
  const int lane = tid & 31;
  const int hi = lane >> 4;
  const int l = lane & 15;
  const int bm = blockIdx.x >> 4;                // 0..124
  const int tn = blockIdx.x & 15;                // 0..15
  const int tm = bm * 5 + wid;                   // 0..624, exact
  const int jbase = tn * 16;

  // Stage B panel: 256 rows x 16 cols, 4 floats per async b128 transfer.
  for (int idx = tid; idx < DIM * 4; idx += 160) {
    const int k = idx >> 2, c4 = idx & 3;
    async_copy_b128(&Bs[k * 16 + c4 * 4], B + (size_t)k * DIM + jbase + c4 * 4);
  }
  async_wait0();
  __syncthreads();

  const float* Arow = A + (size_t)(tm * 16 + l) * DIM;
  v8f acc = {};
  for (int k = 0; k < DIM; k += 4) {
    const int kk = k + 2 * hi;
    v2f a, b;
    a.x = Arow[kk];
    a.y = Arow[kk + 1];
    b.x = Bs[kk * 16 + l];
    b.y = Bs[(kk + 1) * 16 + l];
    acc = wmma_f32(a, b, acc);
  }

  float* Cc = C + jbase + l;
#pragma unroll
  for (int r = 0; r < 8; ++r)
    Cc[(size_t)(tm * 16 + r + 8 * hi) * DIM] = acc[r];
}

__global__ __launch_bounds__(256) void zero_buf(float* __restrict__ p) {
  p[(size_t)blockIdx.x * 256 + threadIdx.x] = 0.0f;
}

// agg[dst[e], :] += m[src[e], :] * attr[e]; thread t owns feature t, 8 edges/block.
__global__ __launch_bounds__(256) void scatter_add(const float* __restrict__ m,
                                                   const int* __restrict__ src,
                                                   const int* __restrict__ dst,
                                                   const float* __restrict__ attr,
                                                   float* __restrict__ agg) {
  const int t = threadIdx.x;
  const int e0 = blockIdx.x * 8;
#pragma unroll
  for (int i = 0; i < 8; ++i) {
    const int e = e0 + i;
    const int s = src[e];
    const int d = dst[e];
    const float w = attr[e];
    atomicAdd(&agg[(size_t)d * DIM + t], m[(size_t)s * DIM + t] * w);
  }
}

// ---------------------------------------------------------------------------
// Fused GRU cell. 5 waves/block share one 16-col slice of all six transposed
// weight panels (w_ih r/z/n, w_hh r/z/n), staged K-chunk at a time via async
// b128 copies into padded LDS (6*16*68*4B = 25.5 KB). Cuts weight L2 traffic
// 5x and turns the 1KB-strided W^T fragment reads into bank-conflict-free ds
// loads. 6 WMMA accumulators, then per-register gate math.
// ---------------------------------------------------------------------------
__global__ __launch_bounds__(160) void gru_fused(const float* __restrict__ agg,
                                                 const float* __restrict__ h,
                                                 const float* __restrict__ w_ih,
                                                 const float* __restrict__ w_hh,
                                                 const float* __restrict__ b_ih,
                                                 const float* __restrict__ b_hh,
                                                 float* __restrict__ hnext) {
  __shared__ __align__(16) float Ws[6 * 16 * BSTRIDE];
  const int tid = threadIdx.x;
  const int wid = tid >> 5;
  const int lane = tid & 31;
  const int hi = lane >> 4;
  const int l = lane & 15;
  const int bm = blockIdx.x >> 4;
  const int tn = blockIdx.x & 15;
  const int tm = bm * 5 + wid;
  const int jbase = tn * 16;
  const int jcol = jbase + l;

  const float* Wb[6] = {
      w_ih + (size_t)jbase * DIM,
      w_ih + (size_t)(DIM + jbase) * DIM,
      w_ih + (size_t)(2 * DIM + jbase) * DIM,
      w_hh + (size_t)jbase * DIM,
      w_hh + (size_t)(DIM + jbase) * DIM,
      w_hh + (size_t)(2 * DIM + jbase) * DIM};

  const float* Ar = agg + (size_t)(tm * 16 + l) * DIM;
  const float* Hr = h + (size_t)(tm * 16 + l) * DIM;

  v8f acc[6] = {};
  for (int kc = 0; kc < DIM; kc += KC) {
    // Stage: 6 mats x 16 cols x 64 k-values = 1536 b128 transfers / 160 thr.
    for (int idx = tid; idx < 6 * 16 * (KC / 4); idx += 160) {
      const int mat = idx >> 8;              // / (16 * 16)
      const int rem = idx & 255;
      const int c = rem >> 4;
      const int k4 = rem & 15;
      async_copy_b128(&Ws[(mat * 16 + c) * BSTRIDE + k4 * 4],
                      Wb[mat] + (size_t)c * DIM + kc + k4 * 4);
    }
    async_wait0();
    __syncthreads();

    for (int k = 0; k < KC; k += 4) {
      const int kk = k + 2 * hi;
      v2f a, ah;
      a.x  = Ar[kc + kk];  a.y  = Ar[kc + kk + 1];
      ah.x = Hr[kc + kk];  ah.y = Hr[kc + kk + 1];
#pragma unroll
      for (int mat = 0; mat < 6; ++mat) {
        const float* bp = &Ws[(mat * 16 + l) * BSTRIDE + kk];
        v2f b;
        b.x = bp[0];
        b.y = bp[1];
        acc[mat] = wmma_f32(mat < 3 ? a : ah, b, acc[mat]);
      }
    }
    __syncthreads();
  }

  const float bir  = b_ih[jcol];
  const float biz  = b_ih[DIM + jcol];
  const float bin_ = b_ih[2 * DIM + jcol];
  const float bhr  = b_hh[jcol];
  const float bhz  = b_hh[DIM + jcol];
  const float bhn  = b_hh[2 * DIM + jcol];

#pragma unroll
  for (int r = 0; r < 8; ++r) {
    const int row = tm * 16 + r + 8 * hi;
    const float rg = 1.0f / (1.0f + __expf(-(acc[0][r] + bir + acc[3][r] + bhr)));
    const float zg = 1.0f / (1.0f + __expf(-(acc[1][r] + biz + acc[4][r] + bhz)));
    const float ng = tanhf(acc[2][r] + bin_ + rg * (acc[5][r] + bhn));
    const float hold = h[(size_t)row * DIM + jcol];
    hnext[(size_t)row * DIM + jcol] = (1.0f - zg) * ng + zg * hold;
  }
}

extern "C" void kernel_launch(void* const* d_in, const int* in_sizes, int n_in,
                              void* d_out, int out_size, void* d_ws, size_t ws_size,
                              hipStream_t stream) {
  const float* x      = (const float*)d_in[0];
  const int*   ei     = (const int*)d_in[1];   // [2, E] (JAX int64 -> int32 w/o x64)
  const float* attr   = (const float*)d_in[2];
  const float* weight = (const float*)d_in[3]; // [L, D, D]
  const float* w_ih   = (const float*)d_in[4]; // [3D, D]
  const float* w_hh   = (const float*)d_in[5]; // [3D, D]
  const float* b_ih   = (const float*)d_in[6];
  const float* b_hh   = (const float*)d_in[7];
  const int* src = ei;
  const int* dst = ei + NE;

  const size_t nd = (size_t)NN * DIM;
  float* m    = (float*)d_ws;
  float* agg  = m + nd;
  float* hA   = agg + nd;
  float* hB   = hA + nd;
  float* hout = (float*)d_out;

  const float* hprev = x;
  for (int lyr = 0; lyr < LAYERS; ++lyr) {
    float* hn = (lyr == LAYERS - 1) ? hout : ((lyr & 1) ? hB : hA);

    gemm_nn<<<2000, 160, 0, stream>>>(hprev, weight + (size_t)lyr * DIM * DIM, m);
    zero_buf<<<(NN * DIM) / 256, 256, 0, stream>>>(agg);
    scatter_add<<<NE / 8, 256, 0, stream>>>(m, src, dst, attr, agg);
    gru_fused<<<2000, 160, 0, stream>>>(agg, hprev, w_ih, w_hh, b_ih, b_hh, hn);

    hprev = hn;
  }
}